// CausalPropagationAdjacency_16054587752781
// MI455X (gfx1250) — compile-verified
//
#include <hip/hip_runtime.h>
#include <math.h>

// Problem constants (match reference)
#define B_ 4
#define T_ 12
#define N_ 512
#define D_ 128
#define L_ 4
#define H_ 64
#define THRESH 0.1f

typedef __attribute__((ext_vector_type(2))) float v2f;
typedef __attribute__((ext_vector_type(8))) float v8f;

#if defined(__has_builtin)
#if __has_builtin(__builtin_amdgcn_wmma_f32_16x16x4_f32)
#define HAVE_WMMA_F32X4 1
#endif
#endif

// ---------------------------------------------------------------------------
// 32x32 fp32 output block per wave: 2x2 tiles of V_WMMA_F32_16X16X4_F32.
// Per K=4 step: 2 A fragments + 2 B fragments feed 4 WMMAs (1 load/WMMA).
// Fragment layouts per CDNA5 ISA 7.12.2:
//   A 16x4:  lanes 0-15 hold M=lane, K={0,1}; lanes 16-31 K={2,3}
//   B 4x16:  lanes 0-15 hold N=lane, K={0,1}; lanes 16-31 N=lane-16, K={2,3}
//   C/D:     VGPR r -> (M=r, N=lane) lanes 0-15; (M=r+8, N=lane-16) upper half
// acc[mi*2+ni] covers rows mi*16..+15, cols ni*16..+15 of the 32x32 block.
// ---------------------------------------------------------------------------
__device__ __forceinline__ void wmma_block_2x2(const float* __restrict__ A, int lda,
                                               const float* __restrict__ Bm, int ldb,
                                               int K, v8f acc[4]) {
  const int lane = threadIdx.x & 31;
  const int hi   = lane >> 4;
  const int lm   = lane & 15;
  acc[0] = {}; acc[1] = {}; acc[2] = {}; acc[3] = {};
#if HAVE_WMMA_F32X4
  const float* ap0 = A + (size_t)lm * lda + 2 * hi;      // rows 0..15
  const float* ap1 = ap0 + (size_t)16 * lda;             // rows 16..31
  const float* bp0 = Bm + (size_t)(2 * hi) * ldb + lm;   // cols 0..15
  const float* bp1 = bp0 + 16;                           // cols 16..31
  for (int k = 0; k < K; k += 4) {
    v2f a0, a1, b0, b1;
    a0.x = ap0[0]; a0.y = ap0[1];
    a1.x = ap1[0]; a1.y = ap1[1];
    b0.x = bp0[0]; b0.y = bp0[ldb];
    b1.x = bp1[0]; b1.y = bp1[ldb];
    acc[0] = __builtin_amdgcn_wmma_f32_16x16x4_f32(false, a0, false, b0, (short)0, acc[0], false, false);
    acc[1] = __builtin_amdgcn_wmma_f32_16x16x4_f32(false, a0, false, b1, (short)0, acc[1], false, false);
    acc[2] = __builtin_amdgcn_wmma_f32_16x16x4_f32(false, a1, false, b0, (short)0, acc[2], false, false);
    acc[3] = __builtin_amdgcn_wmma_f32_16x16x4_f32(false, a1, false, b1, (short)0, acc[3], false, false);
    ap0 += 4; ap1 += 4;
    bp0 += 4 * (size_t)ldb; bp1 += 4 * (size_t)ldb;
  }
#else
  // Scalar fallback (keeps compiling if the builtin is absent).
  for (int mi = 0; mi < 2; ++mi)
    for (int ni = 0; ni < 2; ++ni)
      for (int r = 0; r < 8; ++r) {
        const int row = mi * 16 + hi * 8 + r;
        const int col = ni * 16 + lm;
        float s = 0.0f;
        for (int k = 0; k < K; ++k)
          s += A[(size_t)row * lda + k] * Bm[(size_t)k * ldb + col];
        acc[mi * 2 + ni][r] = s;
      }
#endif
}

// ---------------------------------------------------------------------------
// Kernel 1: lag-encoder hidden layer.
// h[b,n,l*H+j] = relu(x[b,T-1-l,n,:] @ W1[l] + b1[l]); 512x64, K=128 per (b,l).
// ---------------------------------------------------------------------------
__global__ void lag_hidden_kernel(const float* __restrict__ x,
                                  const float* __restrict__ W1,
                                  const float* __restrict__ b1,
                                  float* __restrict__ h_ws) {
  const int tj = blockIdx.x;            // H/32 = 2
  const int ti = blockIdx.y;            // N/32 = 16
  const int z  = blockIdx.z;            // B*L = 16
  const int b  = z / L_;
  const int l  = z % L_;
  const int t  = T_ - 1 - l;

  const float* A  = x  + (((size_t)b * T_ + t) * N_ + (size_t)ti * 32) * D_;
  const float* Bm = W1 + (size_t)l * D_ * H_ + (size_t)tj * 32;
  v8f acc[4];
  wmma_block_2x2(A, D_, Bm, H_, D_, acc);

  const int lane = threadIdx.x & 31;
  const int hi   = lane >> 4;
  const int col  = lane & 15;
#pragma unroll
  for (int mi = 0; mi < 2; ++mi)
#pragma unroll
    for (int ni = 0; ni < 2; ++ni) {
      const int jc = tj * 32 + ni * 16 + col;
      const float bias = b1[l * H_ + jc];
      float* C = h_ws + ((size_t)b * N_ + ti * 32 + mi * 16 + hi * 8) * (L_ * H_) + l * H_ + jc;
#pragma unroll
      for (int r = 0; r < 8; ++r)
        C[(size_t)r * (L_ * H_)] = fmaxf(acc[mi * 2 + ni][r] + bias, 0.0f);
    }
}

// ---------------------------------------------------------------------------
// Kernel 2: agg[b] = 0.25 * (Hcat (512x256) @ W2flat (256x128) + sum_l b2[l])
// (W2 (L,H,D) row-major flattens exactly to (L*H, D) row-major.)
// ---------------------------------------------------------------------------
__global__ void agg_kernel(const float* __restrict__ h_ws,
                           const float* __restrict__ W2,
                           const float* __restrict__ b2,
                           float* __restrict__ agg) {
  const int tj = blockIdx.x;            // D/32 = 4
  const int ti = blockIdx.y;            // 16
  const int b  = blockIdx.z;            // 4

  const float* A  = h_ws + ((size_t)b * N_ + ti * 32) * (L_ * H_);
  const float* Bm = W2 + (size_t)tj * 32;
  v8f acc[4];
  wmma_block_2x2(A, L_ * H_, Bm, D_, L_ * H_, acc);

  const int lane = threadIdx.x & 31;
  const int hi   = lane >> 4;
  const int col  = lane & 15;
#pragma unroll
  for (int mi = 0; mi < 2; ++mi)
#pragma unroll
    for (int ni = 0; ni < 2; ++ni) {
      const int c = tj * 32 + ni * 16 + col;
      const float sb2 = b2[0 * D_ + c] + b2[1 * D_ + c] + b2[2 * D_ + c] + b2[3 * D_ + c];
      float* C = agg + ((size_t)b * N_ + ti * 32 + mi * 16 + hi * 8) * D_ + c;
#pragma unroll
      for (int r = 0; r < 8; ++r)
        C[(size_t)r * D_] = 0.25f * (acc[mi * 2 + ni][r] + sb2);
    }
}

// ---------------------------------------------------------------------------
// Kernel 3: src/tgt projections.  proj[b,p] = agg[b] @ Ws1[p*D:(p+1)*D, :]
// ---------------------------------------------------------------------------
__global__ void proj_kernel(const float* __restrict__ agg,
                            const float* __restrict__ Ws1,
                            float* __restrict__ proj) {
  const int tj = blockIdx.x;            // 4
  const int ti = blockIdx.y;            // 16
  const int z  = blockIdx.z;            // B*2 = 8
  const int b  = z >> 1;
  const int p  = z & 1;

  const float* A  = agg + ((size_t)b * N_ + ti * 32) * D_;
  const float* Bm = Ws1 + (size_t)p * D_ * D_ + (size_t)tj * 32;
  v8f acc[4];
  wmma_block_2x2(A, D_, Bm, D_, D_, acc);

  const int lane = threadIdx.x & 31;
  const int hi   = lane >> 4;
  const int col  = lane & 15;
#pragma unroll
  for (int mi = 0; mi < 2; ++mi)
#pragma unroll
    for (int ni = 0; ni < 2; ++ni) {
      float* C = proj + (((size_t)b * 2 + p) * N_ + ti * 32 + mi * 16 + hi * 8) * D_
                 + tj * 32 + ni * 16 + col;
#pragma unroll
      for (int r = 0; r < 8; ++r)
        C[(size_t)r * D_] = acc[mi * 2 + ni][r];
    }
}

// ---------------------------------------------------------------------------
// Kernel 4: pairwise scorer (ReLU inside the reduction -> VALU, LDS tiled).
// adj[b,i,j] = thresholded sigmoid(sum_d relu(src[i,d]+tgt[j,d]+bs1[d]) * w2[d] + bs2)
// ---------------------------------------------------------------------------
__global__ void score_kernel(const float* __restrict__ proj,
                             const float* __restrict__ bs1,
                             const float* __restrict__ Ws2,
                             const float* __restrict__ bs2,
                             float* __restrict__ adj) {
  __shared__ float s_src[32 * D_];
  __shared__ float s_tgt[32 * D_];
  __shared__ float s_w2[D_];

  const int b  = blockIdx.z;
  const int i0 = blockIdx.y * 32;
  const int j0 = blockIdx.x * 32;
  const int tid = threadIdx.x;

  const float* src = proj + ((size_t)b * 2 + 0) * N_ * D_;
  const float* tgt = proj + ((size_t)b * 2 + 1) * N_ * D_;

  for (int idx = tid; idx < 32 * D_; idx += 256) {
    const int r = idx >> 7;
    const int d = idx & (D_ - 1);
    s_src[idx] = src[(size_t)(i0 + r) * D_ + d];
    s_tgt[idx] = tgt[(size_t)(j0 + r) * D_ + d] + bs1[d];  // fold bs1 into tgt
  }
  for (int idx = tid; idx < D_; idx += 256) s_w2[idx] = Ws2[idx];
  __syncthreads();

  const float c2 = bs2[0];
#pragma unroll
  for (int q = 0; q < 4; ++q) {
    const int il = (tid >> 5) + q * 8;
    const int jl = tid & 31;
    const float* sp = s_src + il * D_;
    const float* tp = s_tgt + jl * D_;
    float s = c2;
    for (int d = 0; d < D_; ++d)
      s += fmaxf(sp[d] + tp[d], 0.0f) * s_w2[d];
    const float sig = 1.0f / (1.0f + __expf(-s));
    float a = (sig > THRESH) ? sig : 0.0f;
    const int gi = i0 + il, gj = j0 + jl;
    if (gi == gj) a = 0.0f;
    adj[((size_t)b * N_ + gi) * N_ + gj] = a;
  }
}

// ---------------------------------------------------------------------------
// Kernel 5: a2 = adj @ adj  (512^3 fp32, WMMA, 32x32 block per wave)
// ---------------------------------------------------------------------------
__global__ void mm_a2_kernel(const float* __restrict__ adj, float* __restrict__ a2) {
  const int tj = blockIdx.x;            // 16
  const int ti = blockIdx.y;            // 16
  const int b  = blockIdx.z;
  const float* base = adj + (size_t)b * N_ * N_;
  v8f acc[4];
  wmma_block_2x2(base + (size_t)ti * 32 * N_, N_, base + (size_t)tj * 32, N_, N_, acc);

  const int lane = threadIdx.x & 31;
  const int hi   = lane >> 4;
  const int col  = lane & 15;
#pragma unroll
  for (int mi = 0; mi < 2; ++mi)
#pragma unroll
    for (int ni = 0; ni < 2; ++ni) {
      float* C = a2 + (size_t)b * N_ * N_ + ((size_t)ti * 32 + mi * 16 + hi * 8) * N_
                 + tj * 32 + ni * 16 + col;
#pragma unroll
      for (int r = 0; r < 8; ++r)
        C[(size_t)r * N_] = acc[mi * 2 + ni][r];
    }
}

// ---------------------------------------------------------------------------
// Kernel 6: a3 = a2 @ adj fused with enhanced = adj + 0.5*a2 + 0.25*a3,
// plus per-batch running max (values >= 0 -> uint atomicMax on float bits).
// ---------------------------------------------------------------------------
__global__ void mm_a3_enh_kernel(const float* __restrict__ a2,
                                 const float* __restrict__ adj,
                                 float* __restrict__ out,
                                 float* __restrict__ maxbuf) {
  const int tj = blockIdx.x;
  const int ti = blockIdx.y;
  const int b  = blockIdx.z;
  const float* a2b  = a2  + (size_t)b * N_ * N_;
  const float* adjb = adj + (size_t)b * N_ * N_;
  v8f acc[4];
  wmma_block_2x2(a2b + (size_t)ti * 32 * N_, N_, adjb + (size_t)tj * 32, N_, N_, acc);

  const int lane = threadIdx.x & 31;
  const int hi   = lane >> 4;
  const int col  = lane & 15;
  float m = 0.0f;
#pragma unroll
  for (int mi = 0; mi < 2; ++mi)
#pragma unroll
    for (int ni = 0; ni < 2; ++ni) {
      const size_t off0 = ((size_t)ti * 32 + mi * 16 + hi * 8) * N_ + tj * 32 + ni * 16 + col;
#pragma unroll
      for (int r = 0; r < 8; ++r) {
        const size_t off = off0 + (size_t)r * N_;
        const float e = adjb[off] + 0.5f * a2b[off] + 0.25f * acc[mi * 2 + ni][r];
        out[(size_t)b * N_ * N_ + off] = e;
        m = fmaxf(m, e);
      }
    }
  atomicMax(reinterpret_cast<unsigned int*>(maxbuf) + b, __float_as_uint(m));
}

// ---------------------------------------------------------------------------
// Kernel 7a: zero the per-batch max buffer (graph-capture safe).
// Kernel 7b: normalize out by (max + 1e-8).
// ---------------------------------------------------------------------------
__global__ void init_max_kernel(float* __restrict__ maxbuf) {
  if (threadIdx.x < B_) maxbuf[threadIdx.x] = 0.0f;
}

__global__ void norm_kernel(float* __restrict__ out, const float* __restrict__ maxbuf) {
  const size_t idx = (size_t)blockIdx.x * 256 + threadIdx.x;
  const size_t total = (size_t)B_ * N_ * N_;
  if (idx < total) {
    const int b = (int)(idx / ((size_t)N_ * N_));
    out[idx] = out[idx] / (maxbuf[b] + 1e-8f);
  }
}

// ---------------------------------------------------------------------------
extern "C" void kernel_launch(void* const* d_in, const int* in_sizes, int n_in,
                              void* d_out, int out_size, void* d_ws, size_t ws_size,
                              hipStream_t stream) {
  (void)in_sizes; (void)n_in; (void)out_size; (void)ws_size;

  const float* x   = (const float*)d_in[0];
  const float* W1  = (const float*)d_in[1];
  const float* b1  = (const float*)d_in[2];
  const float* W2  = (const float*)d_in[3];
  const float* b2  = (const float*)d_in[4];
  const float* Ws1 = (const float*)d_in[5];
  const float* bs1 = (const float*)d_in[6];
  const float* Ws2 = (const float*)d_in[7];
  const float* bs2 = (const float*)d_in[8];
  float* out = (float*)d_out;

  // Workspace layout (floats). Total ~13.6 MB.
  float* w      = (float*)d_ws;
  float* h_ws   = w;                                   // B*N*L*H = 524288
  float* agg    = h_ws + (size_t)B_ * N_ * L_ * H_;    // B*N*D   = 262144
  float* proj   = agg  + (size_t)B_ * N_ * D_;         // B*2*N*D = 524288
  float* adj    = proj + (size_t)B_ * 2 * N_ * D_;     // B*N*N   = 1048576
  float* a2     = adj  + (size_t)B_ * N_ * N_;         // B*N*N   = 1048576
  float* maxbuf = a2   + (size_t)B_ * N_ * N_;         // B

  init_max_kernel<<<1, 32, 0, stream>>>(maxbuf);

  lag_hidden_kernel<<<dim3(H_ / 32, N_ / 32, B_ * L_), 32, 0, stream>>>(x, W1, b1, h_ws);
  agg_kernel<<<dim3(D_ / 32, N_ / 32, B_), 32, 0, stream>>>(h_ws, W2, b2, agg);
  proj_kernel<<<dim3(D_ / 32, N_ / 32, B_ * 2), 32, 0, stream>>>(agg, Ws1, proj);
  score_kernel<<<dim3(N_ / 32, N_ / 32, B_), 256, 0, stream>>>(proj, bs1, Ws2, bs2, adj);
  mm_a2_kernel<<<dim3(N_ / 32, N_ / 32, B_), 32, 0, stream>>>(adj, a2);
  mm_a3_enh_kernel<<<dim3(N_ / 32, N_ / 32, B_), 32, 0, stream>>>(a2, adj, out, maxbuf);

  const size_t total = (size_t)B_ * N_ * N_;
  norm_kernel<<<(unsigned)((total + 255) / 256), 256, 0, stream>>>(out, maxbuf);
}